// CTCLayer_38482906972873
// MI455X (gfx1250) — compile-verified
//
#include <hip/hip_runtime.h>
#include <stdint.h>

// ---------------------------------------------------------------------------
// CTC negative-log-likelihood, fused single pass on gfx1250 (MI455X).
//   kernel 1: one wave32 per batch element.
//     - TDM (tensor_load_to_lds) streams [32 x 96] f32 prediction tiles into
//       double-buffered LDS, synced with s_wait_tensorcnt; DMA of chunk k+1
//       overlaps compute of chunk k.
//     - per-row logsumexp over C=96 with 3 values/lane + shuffle butterflies.
//     - 97-state alpha recursion kept ENTIRELY in registers (4 states/lane);
//       s-1 / s-2 neighbors via __shfl_up + cross-chunk lane patch-up.
//       Zero barriers and zero LDS traffic on the recursion chain.
//   kernel 2: mean over B per-example losses.
// ---------------------------------------------------------------------------

namespace {
constexpr int   kC     = 96;          // classes (blank = kC-1)
constexpr int   kL     = 48;          // max label length
constexpr int   kS     = 2 * kL + 1;  // 97 extended states
constexpr int   kChunk = 32;          // timesteps per TDM tile
constexpr float kNeg   = -1e30f;
}  // namespace

typedef unsigned int v4u __attribute__((ext_vector_type(4)));
typedef int          v4i __attribute__((ext_vector_type(4)));
typedef int          v8i __attribute__((ext_vector_type(8)));

// Device pass: use the Tensor Data Mover unconditionally (compile error here
// would mean the builtin is absent — we want that signal, not a silent
// fallback). Host pass only needs to parse; give it a stub.
#if defined(__HIP_DEVICE_COMPILE__)
#define CTC_USE_TDM 1
#else
#define CTC_USE_TDM 0
#endif

// Issue one TDM load of a [tile_rows x kC] f32 tile (row-major, stride kC)
// from global memory into LDS. Descriptor packing per CDNA5 ISA ch.8.
__device__ __forceinline__ void tdm_rows_to_lds(void* lds_dst, const float* gsrc,
                                                int rows_remaining, int tile_rows) {
#if CTC_USE_TDM
  uint64_t ga = (uint64_t)(uintptr_t)gsrc;
  uint32_t la = (uint32_t)(uintptr_t)lds_dst;  // flat LDS addr truncates to offset

  v4u g0;
  g0.x = 1u;                                                  // count=1, user mode
  g0.y = la;                                                  // lds_addr
  g0.z = (uint32_t)ga;                                        // global_addr[31:0]
  g0.w = (uint32_t)((ga >> 32) & 0x1ffffffull) | (2u << 30);  // addr[56:32] | type=2

  v8i g1;
  g1[0] = (int)(2u << 16);                                    // data_size=4B; mask=0
  g1[1] = (int)((uint32_t)kC << 16);                          // tensor_dim0[15:0]
  g1[2] = (int)(((uint32_t)rows_remaining & 0xffffu) << 16);  // tensor_dim1[15:0]
  g1[3] = (int)(((uint32_t)rows_remaining >> 16) |            // tensor_dim1[31:16]
                ((uint32_t)kC << 16));                        // tile_dim0
  g1[4] = (int)((uint32_t)tile_rows & 0xffffu);               // tile_dim1 (tile_dim2=0)
  g1[5] = kC;                                                 // tensor_dim0_stride lo32
  g1[6] = 0;                                                  // stride hi / dim1_stride lo
  g1[7] = 0;                                                  // dim1_stride hi

  v4i g2;
  g2[0] = 1; g2[1] = 0; g2[2] = 0; g2[3] = 0;                 // tensor_dim2=1, rest 0
  v4i g3;
  g3[0] = 0; g3[1] = 0; g3[2] = 0; g3[3] = 0;

#if defined(__clang_major__) && (__clang_major__ >= 23)
  v8i g4;
  g4[0] = 0; g4[1] = 0; g4[2] = 0; g4[3] = 0;
  g4[4] = 0; g4[5] = 0; g4[6] = 0; g4[7] = 0;
  __builtin_amdgcn_tensor_load_to_lds(g0, g1, g2, g3, g4, 0);
#else
  __builtin_amdgcn_tensor_load_to_lds(g0, g1, g2, g3, 0);
#endif
#else
  (void)lds_dst; (void)gsrc; (void)rows_remaining; (void)tile_rows;
#endif
}

__device__ __forceinline__ void tdm_wait_all() {
#if CTC_USE_TDM
  __builtin_amdgcn_s_wait_tensorcnt(0);
#endif
}

// Uniform-index pick of extended-state value from the 4 per-lane registers.
__device__ __forceinline__ float pick_state(const float a0, const float a1,
                                            const float a2, const float a3, int s) {
  const int k = s >> 5;  // uniform across the wave
  float v = a0;
  if (k == 1) v = a1;
  else if (k == 2) v = a2;
  else if (k == 3) v = a3;
  return __shfl(v, s & 31, 32);
}

__global__ __launch_bounds__(32) void ctc_forward_kernel(
    const int* __restrict__ labels, const float* __restrict__ pred,
    float* __restrict__ loss, int T) {
  __shared__ float tile[2][kChunk * kC];  // double-buffered prediction rows
  __shared__ int   extl[kS];              // extended label sequence
  __shared__ int   labs[kL];              // clamped labels

  const int b    = (int)blockIdx.x;
  const int lane = (int)threadIdx.x;
  const float* base = pred + (size_t)b * (size_t)T * (size_t)kC;

  // ---- label / extended-sequence setup (once; barriers OK here) ----
  int cnt = 0;
  for (int i = lane; i < kL; i += 32) {
    int lv = labels[(size_t)b * kL + i];
    cnt += (lv != -1) ? 1 : 0;
    labs[i] = lv < 0 ? 0 : lv;
  }
  for (int off = 16; off; off >>= 1) cnt += __shfl_xor(cnt, off, 32);
  const int label_len = cnt;
  __syncthreads();
  for (int s = lane; s < kS; s += 32) extl[s] = (s & 1) ? labs[s >> 1] : (kC - 1);
  __syncthreads();

  int  exts[4];
  bool al2[4], vld[4];
#pragma unroll
  for (int k = 0; k < 4; ++k) {
    const int s = lane + 32 * k;
    const bool act = (s < kS);
    const int e = act ? extl[s] : 0;
    exts[k] = e;
    al2[k]  = act && (s & 1) && ((s < 3) || (e != extl[s - 2]));
    vld[k]  = act && (s < 2 * label_len + 1);
  }

  // ---- stream prediction tiles via TDM, double buffered ----
  const int nch = (T + kChunk - 1) / kChunk;
  tdm_rows_to_lds(&tile[0][0], base, T, min(kChunk, T));
  tdm_wait_all();
  if (nch > 1)
    tdm_rows_to_lds(&tile[1][0], base + (size_t)kChunk * kC, T - kChunk,
                    min(kChunk, T - kChunk));

  // alpha for own 4 states, register-resident across all T steps
  float a0 = kNeg, a1r = kNeg, a2r = kNeg, a3r = kNeg;

  int t = 0;
  for (int ch = 0; ch < nch; ++ch) {
    const float* buf = &tile[ch & 1][0];
    const int nr = min(kChunk, T - ch * kChunk);

    for (int r = 0; r < nr; ++r, ++t) {
      const float* row = buf + r * kC;

      // logsumexp over 96 classes: 3 values/lane + shuffle butterflies
      const float x0 = row[lane], x1 = row[lane + 32], x2 = row[lane + 64];
      float m = fmaxf(x0, fmaxf(x1, x2));
      for (int off = 16; off; off >>= 1) m = fmaxf(m, __shfl_xor(m, off, 32));
      float es = __expf(x0 - m) + __expf(x1 - m) + __expf(x2 - m);
      for (int off = 16; off; off >>= 1) es += __shfl_xor(es, off, 32);
      const float lse = m + __logf(es);

      // emission log-probs for own states (independent of alpha chain)
      const float lp0 = row[exts[0]] - lse;
      const float lp1 = row[exts[1]] - lse;
      const float lp2 = row[exts[2]] - lse;
      const float lp3 = row[exts[3]] - lse;

      if (t == 0) {
        // alpha0: only s==0 (blank) and s==1 (first label, if any)
        a0 = (lane == 0) ? lp0 : ((lane == 1 && vld[0]) ? lp0 : kNeg);
        a1r = kNeg; a2r = kNeg; a3r = kNeg;
      } else {
        float old0 = a0, old1 = a1r, old2 = a2r, old3 = a3r;

        // s-1 neighbors: shuffle-up by 1, lane 0 patches from prev chunk lane 31
        float u1_0 = __shfl_up(old0, 1, 32);
        float u1_1 = __shfl_up(old1, 1, 32);
        float u1_2 = __shfl_up(old2, 1, 32);
        float u1_3 = __shfl_up(old3, 1, 32);
        const float b31_0 = __shfl(old0, 31, 32);
        const float b31_1 = __shfl(old1, 31, 32);
        const float b31_2 = __shfl(old2, 31, 32);
        if (lane == 0) { u1_0 = kNeg; u1_1 = b31_0; u1_2 = b31_1; u1_3 = b31_2; }

        // s-2 neighbors: shuffle-up by 2, lanes 0..1 patch from prev chunk 30+lane
        float u2_0 = __shfl_up(old0, 2, 32);
        float u2_1 = __shfl_up(old1, 2, 32);
        float u2_2 = __shfl_up(old2, 2, 32);
        float u2_3 = __shfl_up(old3, 2, 32);
        const float c_0 = __shfl(old0, 30 + lane, 32);
        const float c_1 = __shfl(old1, 30 + lane, 32);
        const float c_2 = __shfl(old2, 30 + lane, 32);
        if (lane < 2) { u2_0 = kNeg; u2_1 = c_0; u2_2 = c_1; u2_3 = c_2; }

        // 3-way logaddexp + emission per state
        {
          const float y3 = al2[0] ? u2_0 : kNeg;
          const float mm = fmaxf(old0, fmaxf(u1_0, y3));
          const float sm = __expf(old0 - mm) + __expf(u1_0 - mm) + __expf(y3 - mm);
          a0 = vld[0] ? (mm + __logf(sm) + lp0) : kNeg;
        }
        {
          const float y3 = al2[1] ? u2_1 : kNeg;
          const float mm = fmaxf(old1, fmaxf(u1_1, y3));
          const float sm = __expf(old1 - mm) + __expf(u1_1 - mm) + __expf(y3 - mm);
          a1r = vld[1] ? (mm + __logf(sm) + lp1) : kNeg;
        }
        {
          const float y3 = al2[2] ? u2_2 : kNeg;
          const float mm = fmaxf(old2, fmaxf(u1_2, y3));
          const float sm = __expf(old2 - mm) + __expf(u1_2 - mm) + __expf(y3 - mm);
          a2r = vld[2] ? (mm + __logf(sm) + lp2) : kNeg;
        }
        {
          const float y3 = al2[3] ? u2_3 : kNeg;
          const float mm = fmaxf(old3, fmaxf(u1_3, y3));
          const float sm = __expf(old3 - mm) + __expf(u1_3 - mm) + __expf(y3 - mm);
          a3r = vld[3] ? (mm + __logf(sm) + lp3) : kNeg;
        }
      }
    }

    // chunk ch+1 is in flight; wait for it, then prefetch ch+2 into this buffer
    if (ch + 1 < nch) {
      tdm_wait_all();
      if (ch + 2 < nch) {
        const int st = (ch + 2) * kChunk;
        tdm_rows_to_lds(&tile[ch & 1][0], base + (size_t)st * kC, T - st,
                        min(kChunk, T - st));
      }
    }
  }

  // ---- final: logaddexp of alpha[2*len] (blank) and alpha[2*len-1] (label) ----
  const int i1 = 2 * label_len;
  const float e1  = pick_state(a0, a1r, a2r, a3r, i1);
  const float e2p = pick_state(a0, a1r, a2r, a3r, i1 > 0 ? i1 - 1 : 0);
  if (lane == 0) {
    const float e2 = (label_len > 0) ? e2p : kNeg;
    const float mm = fmaxf(e1, e2);
    const float ll = mm + __logf(__expf(e1 - mm) + __expf(e2 - mm));
    loss[b] = -ll;
  }
}

__global__ __launch_bounds__(256) void mean_reduce_kernel(
    const float* __restrict__ v, float* __restrict__ out, int n) {
  __shared__ float sm[256];
  float s = 0.f;
  for (int i = (int)threadIdx.x; i < n; i += 256) s += v[i];
  sm[threadIdx.x] = s;
  __syncthreads();
  for (int w = 128; w > 0; w >>= 1) {
    if ((int)threadIdx.x < w) sm[threadIdx.x] += sm[threadIdx.x + w];
    __syncthreads();
  }
  if (threadIdx.x == 0) out[0] = sm[0] / (float)n;
}

extern "C" void kernel_launch(void* const* d_in, const int* in_sizes, int n_in,
                              void* d_out, int out_size, void* d_ws, size_t ws_size,
                              hipStream_t stream) {
  (void)n_in; (void)out_size; (void)ws_size;
  const int*   labels = (const int*)d_in[0];   // [B, 48] int32
  const float* pred   = (const float*)d_in[1]; // [B, T, 96] f32
  const int B = in_sizes[0] / kL;
  const int T = in_sizes[1] / (B * kC);
  float* per_b = (float*)d_ws;                 // B floats of scratch

  ctc_forward_kernel<<<B, 32, 0, stream>>>(labels, pred, per_b, T);
  mean_reduce_kernel<<<1, 256, 0, stream>>>(per_b, (float*)d_out, B);
}